// maxpooler_ring_79585743994952
// MI455X (gfx1250) — compile-verified
//
#include <hip/hip_runtime.h>

// ---------------------------------------------------------------------------
// Per-ring Conv1d(64->128,k=1) + training BatchNorm1d + per-(batch,ring) max
// broadcast, for MI455X (gfx1250, wave32, WMMA).
//
// Pipeline:
//   init   : zero counters/sums, init encoded max/min
//   bin    : counting-sort point ids by (batch,ring) segment (u16 perm)
//   gemm   : per-segment WMMA f32 16x16x4 GEMM; accumulate sum/sumsq per
//            (ring,ch) and max/min per (seg,ch) -- y is never stored
//   tab    : finalize BN affine + pick max/min -> broadcast table [128][128]
//   out    : stream table to out[B,128,N] via ring ids (coalesced)
// ---------------------------------------------------------------------------

namespace {
constexpr int kB    = 16;
constexpr int kCin  = 64;
constexpr int kN    = 8192;
constexpr int kCout = 128;
constexpr int kRing = 8;
constexpr int kSeg  = kB * kRing;   // 128 (batch,ring) segments
constexpr int kM    = kB * kN;      // 131072 points
constexpr float kEps = 1e-5f;

// workspace layout in 32-bit words
constexpr int W_CNT = 0;                       // 128 int   per-seg counts
constexpr int W_S   = 128;                     // 1024 f    per-(ring,ch) sum
constexpr int W_SS  = 1152;                    // 1024 f    per-(ring,ch) sumsq
constexpr int W_MX  = 2176;                    // 16384 u32 enc max per (seg,ch)
constexpr int W_MN  = 18560;                   // 16384 u32 enc min per (seg,ch)
constexpr int W_TAB = 34944;                   // 16384 f   broadcast table
constexpr int W_END = W_TAB + kSeg * kCout;    // 51328 words
constexpr size_t W_PERM_BYTES = size_t(W_END) * 4;  // then u16 perm[128][8192]
}

typedef float v2f __attribute__((ext_vector_type(2)));
typedef float v8f __attribute__((ext_vector_type(8)));

// order-preserving float<->uint encoding so max/min can use integer atomics
__device__ __forceinline__ unsigned enc_ord(float f) {
  unsigned u = __float_as_uint(f);
  return (u & 0x80000000u) ? ~u : (u | 0x80000000u);
}
__device__ __forceinline__ float dec_ord(unsigned u) {
  return __uint_as_float((u & 0x80000000u) ? (u & 0x7FFFFFFFu) : ~u);
}

// ---------------------------------------------------------------------------
__global__ void init_kernel(unsigned* __restrict__ ws) {
  int i = blockIdx.x * blockDim.x + threadIdx.x;
  if (i >= W_TAB) return;
  ws[i] = (i < W_MN) ? 0u : 0xFFFFFFFFu;   // cnt/s/ss/mx(enc -inf)=0, mn=~0
}

// ---------------------------------------------------------------------------
__global__ void bin_kernel(const int* __restrict__ ring, int* __restrict__ cnt,
                           unsigned short* __restrict__ perm) {
  int m = blockIdx.x * blockDim.x + threadIdx.x;
  if (m >= kM) return;
  int b = m >> 13, n = m & (kN - 1);
  int r = ring[m] & 7;
  int seg = b * kRing + r;
  int pos = atomicAdd(&cnt[seg], 1);
  perm[(size_t)seg * kN + pos] = (unsigned short)n;
}

// ---------------------------------------------------------------------------
// 128 threads = 4 waves. Block owns one (batch,ring) segment x 256 point slots.
// Wave w owns output channels [32w, 32w+32): two 16x16 WMMA column tiles.
__global__ __launch_bounds__(128)
void ring_gemm_kernel(const float* __restrict__ x, const float* __restrict__ W,
                      const float* __restrict__ bias, const int* __restrict__ cnt,
                      const unsigned short* __restrict__ perm,
                      float* __restrict__ s, float* __restrict__ ss,
                      unsigned* __restrict__ mx, unsigned* __restrict__ mn) {
  const int seg   = blockIdx.x >> 5;
  const int start = (blockIdx.x & 31) * 256;
  const int b = seg >> 3;
  const int r = seg & 7;
  const int cseg = cnt[seg];
  if (start >= cseg) return;            // block-uniform

  const int tid   = threadIdx.x;
  const int lane  = tid & 31;
  const int wave  = tid >> 5;
  const int rowA  = lane & 15;          // M row (A) / N col (B,C)
  const int kh    = lane >> 4;          // lane half: K+0/1 vs K+2/3
  const int cbase = wave * 32;
  const int pbase = 8 * kh;             // C-layout point offset of this half

  __shared__ float xs[16 * 68];         // 16 points x 64 feats, padded stride

  __builtin_prefetch(W + (size_t)r * kCout * kCin, 0, 3);

  // Preload this wave's B-operand (W^T) into registers: 2 tiles x 16 k-steps.
  // B layout (4x16): v0 = {K=k0 | K=k0+2}, v1 = {K=k0+1 | K=k0+3}.
  v2f breg[2][16];
  float bb[2];
  #pragma unroll
  for (int t = 0; t < 2; ++t) {
    const int cout = cbase + t * 16 + rowA;
    const float* wp = W + (size_t)(r * kCout + cout) * kCin + 2 * kh;
    #pragma unroll
    for (int kk = 0; kk < 16; ++kk)
      breg[t][kk] = *(const v2f*)(wp + kk * 4);
    bb[t] = bias[r * kCout + cout];
  }

  float sumA[2] = {0.f, 0.f}, ssA[2] = {0.f, 0.f};
  float mxA[2] = {-3.4e38f, -3.4e38f}, mnA[2] = {3.4e38f, 3.4e38f};

  const float* xb = x + (size_t)b * kCin * kN;
  const unsigned short* pseg = perm + (size_t)seg * kN;

  for (int tile = 0; tile < 16; ++tile) {
    const int tstart = start + tile * 16;
    if (tstart >= cseg) break;          // uniform: cseg same for all threads
    __syncthreads();                    // previous tile's xs reads done

    // Gather 16 points x 64 features (x is [C][N] -> per-lane gather), pad 0.
    #pragma unroll
    for (int i = 0; i < 8; ++i) {
      const int idx = tid + i * 128;    // 0..1023
      const int p = idx >> 6, c = idx & 63;
      float v = 0.f;
      if (tstart + p < cseg) {
        const int nn = pseg[tstart + p];
        v = xb[(size_t)c * kN + nn];
      }
      xs[p * 68 + c] = v;
    }
    __syncthreads();

    const int vt = cseg - tstart;       // valid points in this tile
    #pragma unroll
    for (int t = 0; t < 2; ++t) {
      v8f acc = {};
      #pragma unroll
      for (int kk = 0; kk < 16; ++kk) {
        v2f a;                          // A 16x4: lanes 0-15 rows, K split
        a.x = xs[rowA * 68 + kk * 4 + 2 * kh];
        a.y = xs[rowA * 68 + kk * 4 + 2 * kh + 1];
        acc = __builtin_amdgcn_wmma_f32_16x16x4_f32(
            false, a, false, breg[t][kk], (short)0, acc, false, false);
      }
      // C layout: VGPR i -> point M = i (+8 for hi lanes), channel N = lane%16
      // Branchless masking: invalid rows contribute 0 to sums and neutral
      // values to max/min (v_cndmask instead of exec-branch chains).
      #pragma unroll
      for (int i = 0; i < 8; ++i) {
        const bool ok = (i + pbase) < vt;
        const float y  = acc[i] + bb[t];
        const float ym = ok ? y : 0.f;
        sumA[t] += ym;
        ssA[t]  += ym * ym;
        mxA[t] = fmaxf(mxA[t], ok ? y : -3.4e38f);
        mnA[t] = fminf(mnA[t], ok ? y :  3.4e38f);
      }
    }
  }

  // Merge the two lane halves (points 0-7 vs 8-15 of each channel), commit.
  #pragma unroll
  for (int t = 0; t < 2; ++t) {
    const float su = sumA[t] + __shfl_xor(sumA[t], 16, 32);
    const float sq = ssA[t]  + __shfl_xor(ssA[t], 16, 32);
    const float ma = fmaxf(mxA[t], __shfl_xor(mxA[t], 16, 32));
    const float mi = fminf(mnA[t], __shfl_xor(mnA[t], 16, 32));
    if (lane < 16) {
      const int c = cbase + t * 16 + lane;
      atomicAdd(&s[r * kCout + c], su);
      atomicAdd(&ss[r * kCout + c], sq);
      atomicMax(&mx[seg * kCout + c], enc_ord(ma));
      atomicMin(&mn[seg * kCout + c], enc_ord(mi));
    }
  }
}

// ---------------------------------------------------------------------------
__global__ void tab_kernel(const int* __restrict__ cnt, const float* __restrict__ s,
                           const float* __restrict__ ss, const unsigned* __restrict__ mx,
                           const unsigned* __restrict__ mn, const float* __restrict__ gamma,
                           const float* __restrict__ beta, float* __restrict__ tab) {
  int idx = blockIdx.x * blockDim.x + threadIdx.x;   // seg*128 + c
  if (idx >= kSeg * kCout) return;
  int seg = idx >> 7, c = idx & 127;
  int r = seg & 7;
  float cr = 0.f;
  for (int b2 = 0; b2 < kB; ++b2) cr += (float)cnt[b2 * kRing + r];
  cr = fmaxf(cr, 1.f);
  float mean = s[r * kCout + c] / cr;
  float var  = ss[r * kCout + c] / cr - mean * mean;  // biased, like reference
  float a = gamma[r * kCout + c] * rsqrtf(var + kEps);
  float hi = dec_ord(mx[idx]);
  float lo = dec_ord(mn[idx]);
  float pick = (a >= 0.f) ? hi : lo;   // monotone affine: max(yn)=a*extreme+d
  tab[idx] = a * (pick - mean) + beta[r * kCout + c];
}

// ---------------------------------------------------------------------------
__global__ __launch_bounds__(256)
void out_kernel(const int* __restrict__ ring, const float* __restrict__ tab,
                float* __restrict__ out) {
  const int b = blockIdx.x >> 5;
  const int n = (blockIdx.x & 31) * 256 + threadIdx.x;
  __shared__ float ts[kRing * kCout];
  #pragma unroll
  for (int i = 0; i < 4; ++i) {
    int idx = threadIdx.x + i * 256;
    ts[idx] = tab[b * kRing * kCout + idx];
  }
  __syncthreads();
  const int rr = ring[b * kN + n] & 7;
  float* ob = out + (size_t)b * kCout * kN + n;
  #pragma unroll 4
  for (int c = 0; c < kCout; ++c)
    ob[(size_t)c * kN] = ts[rr * kCout + c];   // coalesced along n
}

// ---------------------------------------------------------------------------
extern "C" void kernel_launch(void* const* d_in, const int* in_sizes, int n_in,
                              void* d_out, int out_size, void* d_ws, size_t ws_size,
                              hipStream_t stream) {
  const float* x     = (const float*)d_in[0];
  const int*   ring  = (const int*)d_in[1];
  const float* W     = (const float*)d_in[2];
  const float* bias  = (const float*)d_in[3];
  const float* gamma = (const float*)d_in[4];
  const float* beta  = (const float*)d_in[5];
  float* out = (float*)d_out;

  unsigned* wsu = (unsigned*)d_ws;
  int*      cnt = (int*)(wsu + W_CNT);
  float*    s   = (float*)(wsu + W_S);
  float*    ss  = (float*)(wsu + W_SS);
  unsigned* mx  = wsu + W_MX;
  unsigned* mn  = wsu + W_MN;
  float*    tab = (float*)(wsu + W_TAB);
  unsigned short* perm = (unsigned short*)((char*)d_ws + W_PERM_BYTES);

  hipLaunchKernelGGL(init_kernel, dim3((W_TAB + 255) / 256), dim3(256), 0, stream, wsu);
  hipLaunchKernelGGL(bin_kernel, dim3(kM / 256), dim3(256), 0, stream, ring, cnt, perm);
  hipLaunchKernelGGL(ring_gemm_kernel, dim3(kSeg * 32), dim3(128), 0, stream,
                     x, W, bias, cnt, perm, s, ss, mx, mn);
  hipLaunchKernelGGL(tab_kernel, dim3((kSeg * kCout + 255) / 256), dim3(256), 0, stream,
                     cnt, s, ss, mx, mn, gamma, beta, tab);
  hipLaunchKernelGGL(out_kernel, dim3(kB * 32), dim3(256), 0, stream, ring, tab, out);
}